// GraphSAGE_51238959841366
// MI455X (gfx1250) — compile-verified
//
#include <hip/hip_runtime.h>
#include <hip/hip_bf16.h>

typedef __attribute__((ext_vector_type(2))) float v2f;
typedef __attribute__((ext_vector_type(8))) float v8f;

#define NNODES 50000
#define NEDGES 600000
#define DIM    128
#define NTILES (NNODES / 16)   // 3125 row tiles of 16

// ---------------------------------------------------------------------------
// Degree accumulation: one thread per edge, hardware f32 atomic into deg[dst].
// unsafeAtomicAdd forces global_atomic_add_f32 (no CAS loop).
// ---------------------------------------------------------------------------
__global__ void deg_kernel(const int* __restrict__ dst, float* __restrict__ deg, int nE) {
  int t = blockIdx.x * blockDim.x + threadIdx.x;
  if (t >= nE) return;
  unsafeAtomicAdd(&deg[dst[t]], 1.0f);
}

// In-place: deg[i] -> 1 / max(deg[i], 1)
__global__ void invdeg_kernel(float* __restrict__ deg, int n) {
  int t = blockIdx.x * blockDim.x + threadIdx.x;
  if (t >= n) return;
  deg[t] = 1.0f / fmaxf(deg[t], 1.0f);
}

// ---------------------------------------------------------------------------
// Edge scatter: one wave (32 lanes) per edge; each lane moves 4 contiguous
// floats with a float4 gather and 4 global_atomic_add_f32 into aggr[dst].
// ---------------------------------------------------------------------------
__global__ void scatter_kernel(const float* __restrict__ x,
                               const int* __restrict__ src,
                               const int* __restrict__ dst,
                               float* __restrict__ aggr, int nE) {
  int t = blockIdx.x * blockDim.x + threadIdx.x;
  int e = t >> 5;
  if (e >= nE) return;
  int lane = t & 31;
  int s = src[e];
  int d = dst[e];
  const float4* xs = (const float4*)(x + (size_t)s * DIM);
  float4 v = xs[lane];                              // lane*4 .. lane*4+3
  float* ap = aggr + (size_t)d * DIM + lane * 4;
  unsafeAtomicAdd(ap + 0, v.x);
  unsafeAtomicAdd(ap + 1, v.y);
  unsafeAtomicAdd(ap + 2, v.z);
  unsafeAtomicAdd(ap + 3, v.w);
}

// ---------------------------------------------------------------------------
// Fused SAGE layer GEMM (fp32 WMMA):
//   out = (aggr * inv_deg) @ Wl + x @ Wr + bias, optional ReLU.
// One wave computes a 16x128 output strip: 8 accumulator tiles (16x16 f32).
// A fragment layout (V_WMMA_F32_16X16X4_F32, 16x4):
//   lanes 0-15 : row = lane,    K = k+0, k+1
//   lanes16-31 : row = lane-16, K = k+2, k+3
// B fragment (4x16): vgpr0 = row K (+0 / +2 per half), vgpr1 = K+1 / K+3.
// C/D: vgpr r -> M = r (lanes 0-15), M = 8+r (lanes 16-31); N = lane&15.
// ---------------------------------------------------------------------------
__global__ void sage_gemm_kernel(const float* __restrict__ xin,
                                 const float* __restrict__ aggr,
                                 const float* __restrict__ invdeg,
                                 const float* __restrict__ Wl,
                                 const float* __restrict__ bl,
                                 const float* __restrict__ Wr,
                                 float* __restrict__ out,
                                 int do_relu) {
  int wave = threadIdx.x >> 5;
  int lane = threadIdx.x & 31;
  int tile = blockIdx.x * (blockDim.x >> 5) + wave;
  if (tile >= NTILES) return;                 // whole-wave uniform exit

  int row0 = tile * 16;
  int half = lane >> 4;                        // 0: K+{0,1}, 1: K+{2,3}
  int l16  = lane & 15;
  int arow = row0 + l16;
  float scale = invdeg[arow];

  const float* aRow = aggr + (size_t)arow * DIM;
  const float* xRow = xin  + (size_t)arow * DIM;

  v8f acc[8];
#pragma unroll
  for (int ct = 0; ct < 8; ++ct) acc[ct] = (v8f){0,0,0,0,0,0,0,0};

  for (int k = 0; k < DIM; k += 4) {
    int ka = k + half * 2;
    v2f aA, aX;
    aA.x = aRow[ka] * scale;
    aA.y = aRow[ka + 1] * scale;
    aX.x = xRow[ka];
    aX.y = xRow[ka + 1];

#pragma unroll
    for (int ct = 0; ct < 8; ++ct) {
      int col = ct * 16 + l16;
      v2f bL;
      bL.x = Wl[(size_t)ka * DIM + col];
      bL.y = Wl[(size_t)(ka + 1) * DIM + col];
      acc[ct] = __builtin_amdgcn_wmma_f32_16x16x4_f32(
          false, aA, false, bL, (short)0, acc[ct], false, false);
    }
#pragma unroll
    for (int ct = 0; ct < 8; ++ct) {
      int col = ct * 16 + l16;
      v2f bR;
      bR.x = Wr[(size_t)ka * DIM + col];
      bR.y = Wr[(size_t)(ka + 1) * DIM + col];
      acc[ct] = __builtin_amdgcn_wmma_f32_16x16x4_f32(
          false, aX, false, bR, (short)0, acc[ct], false, false);
    }
  }

  // Epilogue: bias + optional ReLU, scalar stores.
#pragma unroll
  for (int ct = 0; ct < 8; ++ct) {
    int col = ct * 16 + l16;
    float bias = bl[col];
#pragma unroll
    for (int r = 0; r < 8; ++r) {
      float v = acc[ct][r] + bias;
      if (do_relu) v = fmaxf(v, 0.0f);
      out[(size_t)(row0 + r + half * 8) * DIM + col] = v;
    }
  }
}

// ---------------------------------------------------------------------------
// Host-side orchestration (graph-capture safe: only kernels + MemsetAsync).
// ---------------------------------------------------------------------------
extern "C" void kernel_launch(void* const* d_in, const int* in_sizes, int n_in,
                              void* d_out, int out_size, void* d_ws, size_t ws_size,
                              hipStream_t stream) {
  const float* x   = (const float*)d_in[0];
  const int*   ei  = (const int*)d_in[1];        // [2, E]: row0 = src, row1 = dst
  const float* W_l = (const float*)d_in[2];      // [5, 128, 128]
  const float* b_l = (const float*)d_in[3];      // [5, 128]
  const float* W_r = (const float*)d_in[4];      // [5, 128, 128]
  float* outp = (float*)d_out;

  const int* src = ei;
  const int* dst = ei + NEDGES;

  const size_t feat = (size_t)NNODES * DIM;
  float* buf0 = (float*)d_ws;
  float* buf1 = buf0 + feat;
  float* aggr = buf1 + feat;
  float* deg  = aggr + feat;                     // also holds inv_deg after invert

  // 1) degrees -> inv_deg (recomputed every call: deterministic)
  hipMemsetAsync(deg, 0, NNODES * sizeof(float), stream);
  deg_kernel<<<(NEDGES + 255) / 256, 256, 0, stream>>>(dst, deg, NEDGES);
  invdeg_kernel<<<(NNODES + 255) / 256, 256, 0, stream>>>(deg, NNODES);

  // 2) 5 layers, ping-pong activation buffers
  const float* cur = x;
  for (int layer = 0; layer < 5; ++layer) {
    float* next = (layer == 4) ? outp : ((layer & 1) ? buf1 : buf0);

    hipMemsetAsync(aggr, 0, feat * sizeof(float), stream);
    scatter_kernel<<<(NEDGES * 32 + 255) / 256, 256, 0, stream>>>(
        cur, src, dst, aggr, NEDGES);

    int wavesPerBlock = 8;                       // 256 threads
    int blocks = (NTILES + wavesPerBlock - 1) / wavesPerBlock;
    sage_gemm_kernel<<<blocks, 256, 0, stream>>>(
        cur, aggr, deg,
        W_l + (size_t)layer * DIM * DIM,
        b_l + (size_t)layer * DIM,
        W_r + (size_t)layer * DIM * DIM,
        next, (layer < 4) ? 1 : 0);

    cur = next;
  }
}